// VFNet_60129542144413
// MI455X (gfx1250) — compile-verified
//
#include <hip/hip_runtime.h>
#include <hip/hip_bf16.h>

// ---------------- problem constants (from reference) ----------------
#define NCAM   6
#define NPTS   200000          // 100*100*20 voxels, flattened (z,y,x), x fastest
#define FH     44
#define FW     80
#define MH     352
#define MW     640
#define NFEAT  64
#define C1     65              // 64 feat channels + depth/100
#define EPSF   1e-8f

typedef float v2f __attribute__((ext_vector_type(2)));
typedef float v8f __attribute__((ext_vector_type(8)));

// ---------------------------------------------------------------
// Kernel 1: per camera, M(3x4) = intr(3x3) @ inv(extr)(4x4)[0:3,:]
// general 4x4 inverse via adjugate (matches jnp.linalg.inv numerically
// to fp rounding for these well-conditioned rigid transforms).
// ---------------------------------------------------------------
__global__ void build_mats(const float* __restrict__ intr,
                           const float* __restrict__ extr,
                           float* __restrict__ Mout) {
  int c = threadIdx.x;
  if (c >= NCAM) return;
  const float* m = extr + c * 16;
  float inv[16];
  inv[0]  =  m[5]*m[10]*m[15] - m[5]*m[11]*m[14] - m[9]*m[6]*m[15] + m[9]*m[7]*m[14] + m[13]*m[6]*m[11] - m[13]*m[7]*m[10];
  inv[4]  = -m[4]*m[10]*m[15] + m[4]*m[11]*m[14] + m[8]*m[6]*m[15] - m[8]*m[7]*m[14] - m[12]*m[6]*m[11] + m[12]*m[7]*m[10];
  inv[8]  =  m[4]*m[9]*m[15]  - m[4]*m[11]*m[13] - m[8]*m[5]*m[15] + m[8]*m[7]*m[13] + m[12]*m[5]*m[11] - m[12]*m[7]*m[9];
  inv[12] = -m[4]*m[9]*m[14]  + m[4]*m[10]*m[13] + m[8]*m[5]*m[14] - m[8]*m[6]*m[13] - m[12]*m[5]*m[10] + m[12]*m[6]*m[9];
  inv[1]  = -m[1]*m[10]*m[15] + m[1]*m[11]*m[14] + m[9]*m[2]*m[15] - m[9]*m[3]*m[14] - m[13]*m[2]*m[11] + m[13]*m[3]*m[10];
  inv[5]  =  m[0]*m[10]*m[15] - m[0]*m[11]*m[14] - m[8]*m[2]*m[15] + m[8]*m[3]*m[14] + m[12]*m[2]*m[11] - m[12]*m[3]*m[10];
  inv[9]  = -m[0]*m[9]*m[15]  + m[0]*m[11]*m[13] + m[8]*m[1]*m[15] - m[8]*m[3]*m[13] - m[12]*m[1]*m[11] + m[12]*m[3]*m[9];
  inv[13] =  m[0]*m[9]*m[14]  - m[0]*m[10]*m[13] - m[8]*m[1]*m[14] + m[8]*m[2]*m[13] + m[12]*m[1]*m[10] - m[12]*m[2]*m[9];
  inv[2]  =  m[1]*m[6]*m[15]  - m[1]*m[7]*m[14]  - m[5]*m[2]*m[15] + m[5]*m[3]*m[14] + m[13]*m[2]*m[7]  - m[13]*m[3]*m[6];
  inv[6]  = -m[0]*m[6]*m[15]  + m[0]*m[7]*m[14]  + m[4]*m[2]*m[15] - m[4]*m[3]*m[14] - m[12]*m[2]*m[7]  + m[12]*m[3]*m[6];
  inv[10] =  m[0]*m[5]*m[15]  - m[0]*m[7]*m[13]  - m[4]*m[1]*m[15] + m[4]*m[3]*m[13] + m[12]*m[1]*m[7]  - m[12]*m[3]*m[5];
  inv[14] = -m[0]*m[5]*m[14]  + m[0]*m[6]*m[13]  + m[4]*m[1]*m[14] - m[4]*m[2]*m[13] - m[12]*m[1]*m[6]  + m[12]*m[2]*m[5];
  inv[3]  = -m[1]*m[6]*m[11]  + m[1]*m[7]*m[10]  + m[5]*m[2]*m[11] - m[5]*m[3]*m[10] - m[9]*m[2]*m[7]   + m[9]*m[3]*m[6];
  inv[7]  =  m[0]*m[6]*m[11]  - m[0]*m[7]*m[10]  - m[4]*m[2]*m[11] + m[4]*m[3]*m[10] + m[8]*m[2]*m[7]   - m[8]*m[3]*m[6];
  inv[11] = -m[0]*m[5]*m[11]  + m[0]*m[7]*m[9]   + m[4]*m[1]*m[11] - m[4]*m[3]*m[9]  - m[8]*m[1]*m[7]   + m[8]*m[3]*m[5];
  inv[15] =  m[0]*m[5]*m[10]  - m[0]*m[6]*m[9]   - m[4]*m[1]*m[10] + m[4]*m[2]*m[9]  + m[8]*m[1]*m[6]   - m[8]*m[2]*m[5];
  float det = m[0]*inv[0] + m[1]*inv[4] + m[2]*inv[8] + m[3]*inv[12];
  float rdet = 1.0f / det;
#pragma unroll
  for (int i = 0; i < 16; ++i) inv[i] *= rdet;

  const float* K = intr + c * 9;
  float* Mo = Mout + c * 12;              // row-major 3x4: Mo[n*4+k]
#pragma unroll
  for (int n = 0; n < 3; ++n)
#pragma unroll
    for (int k = 0; k < 4; ++k)
      Mo[n*4 + k] = K[n*3+0]*inv[0*4+k] + K[n*3+1]*inv[1*4+k] + K[n*3+2]*inv[2*4+k];
}

// ---------------------------------------------------------------
// Kernel 2: bilinear resize mask (NCAM,1,352,640) -> (NCAM,1,44,80)
// ---------------------------------------------------------------
__global__ void resize_mask(const float* __restrict__ mask,
                            float* __restrict__ outm) {
  int idx = blockIdx.x * blockDim.x + threadIdx.x;
  if (idx >= NCAM * FH * FW) return;
  int c = idx / (FH * FW);
  int rem = idx - c * (FH * FW);
  int i = rem / FW, j = rem - (rem / FW) * FW;
  float y = (float)i * ((float)(MH - 1) / (float)(FH - 1));   // linspace(0,351,44)
  float x = (float)j * ((float)(MW - 1) / (float)(FW - 1));   // linspace(0,639,80)
  int y0 = (int)floorf(y); int y1 = min(y0 + 1, MH - 1);
  int x0 = (int)floorf(x); int x1 = min(x0 + 1, MW - 1);
  float wy = y - (float)y0, wx = x - (float)x0;
  const float* im = mask + (size_t)c * MH * MW;
  float r0 = im[y0*MW + x0] * (1.f - wy) + im[y1*MW + x0] * wy;
  float r1 = im[y0*MW + x1] * (1.f - wy) + im[y1*MW + x1] * wy;
  outm[idx] = r0 * (1.f - wx) + r1 * wx;
}

// ---------------------------------------------------------------
// Kernel 3: WMMA fp32 projection + ROI + 64-ch bilinear gather.
// grid = (ceil(NPTS/256), NCAM), block = 256 (8 wave32).
// Each wave projects 32 points with two V_WMMA_F32_16X16X4_F32:
//   A (16x4, 2 VGPR/lane): lanes 0-15 hold (x,y), lanes 16-31 hold (z,1)
//   B (4x16, 2 VGPR/lane): B[k][n] = M[n][k], cols 3..15 zero
//   D (16x16 f32, 8 VGPR): row m -> VGPR m&7, lane n (+16 if m>=8)
// ---------------------------------------------------------------
__global__ __launch_bounds__(256) void project_sample(
    const float* __restrict__ Mats, const float* __restrict__ maskr,
    const float* __restrict__ feats, float* __restrict__ out) {
  const int cam  = blockIdx.y;
  const int tid  = threadIdx.x;
  const int lane = tid & 31;
  const int wave = tid >> 5;
  const int n    = blockIdx.x * 256 + tid;        // this thread's point
  const int pbase = blockIdx.x * 256 + wave * 32; // wave's first point
  const float* Mc = Mats + cam * 12;

  // ---- build A for two 16-point WMMAs + shared B ----
  const int half = lane >> 4;        // 0 -> K0,K1 (x,y); 1 -> K2,K3 (z,1)
  const int arow = lane & 15;

  v2f a0, a1, b;
  {
    int q0 = min(pbase + arow,       NPTS - 1);
    int q1 = min(pbase + 16 + arow,  NPTS - 1);
    int xg0 = q0 % 100, t0 = q0 / 100, yg0 = t0 % 100, zg0 = t0 / 100;
    int xg1 = q1 % 100, t1 = q1 / 100, yg1 = t1 % 100, zg1 = t1 / 100;
    float X0 = -50.f + (float)xg0, Y0 = -50.f + (float)yg0, Z0 = -15.f + 1.5f * (float)zg0;
    float X1 = -50.f + (float)xg1, Y1 = -50.f + (float)yg1, Z1 = -15.f + 1.5f * (float)zg1;
    a0.x = half ? Z0 : X0;  a0.y = half ? 1.0f : Y0;
    a1.x = half ? Z1 : X1;  a1.y = half ? 1.0f : Y1;
    int k0 = half ? 2 : 0, k1 = half ? 3 : 1;
    b.x = (arow < 3) ? Mc[arow * 4 + k0] : 0.0f;  // B[k0][arow] = M[arow][k0]
    b.y = (arow < 3) ? Mc[arow * 4 + k1] : 0.0f;
  }

  v8f c0 = {}; v8f c1 = {};
  c0 = __builtin_amdgcn_wmma_f32_16x16x4_f32(false, a0, false, b, (short)0, c0, false, false);
  c1 = __builtin_amdgcn_wmma_f32_16x16x4_f32(false, a1, false, b, (short)0, c1, false, false);

  // ---- extract (px,py,pz) per lane via LDS (only cols 0..2 are real) ----
  __shared__ float lds[8][2][2][3][8];   // [wave][J][half][col n][row r] = D_J[half*8+r][n]
  if (arow < 3) {
#pragma unroll
    for (int r = 0; r < 8; ++r) {
      lds[wave][0][half][arow][r] = c0[r];
      lds[wave][1][half][arow][r] = c1[r];
    }
  }
  __syncthreads();
  const int J  = lane >> 4;          // which WMMA produced my point
  const int m  = lane & 15;          // my row within it
  const int h2 = (m < 8) ? 0 : 1;
  const int r  = m & 7;
  float px = lds[wave][J][h2][0][r];
  float py = lds[wave][J][h2][1][r];
  float pz = lds[wave][J][h2][2][r];

  // ---- validity / ROI ----
  float depth = pz;
  float dinv  = 1.0f / (depth + EPSF);
  float u = px * dinv, v = py * dinv;
  float gx = 2.0f * u / (float)(FW - 1) - 1.0f;
  float gy = 2.0f * v / (float)(FH - 1) - 1.0f;
  bool valid = (gx >= -1.f) && (gx <= 1.f) && (gy >= -1.f) && (gy <= 1.f) && (depth > 0.f);

  float xr = (gx + 1.f) * 0.5f * (float)(FW - 1);
  float yr = (gy + 1.f) * 0.5f * (float)(FH - 1);

  int xi = __float2int_rn(xr);       // round-to-nearest-even == jnp.round
  int yi = __float2int_rn(yr);
  bool inbn = (xi >= 0) && (xi < FW) && (yi >= 0) && (yi < FH);
  int xcn = min(max(xi, 0), FW - 1), ycn = min(max(yi, 0), FH - 1);
  float mval = maskr[cam * FH * FW + ycn * FW + xcn] * (inbn ? 1.f : 0.f);
  float vm = (valid && (mval > 0.5f)) ? 1.0f : 0.0f;

  if (n < NPTS) {
    float* ob = out + (size_t)cam * (C1 * (size_t)NPTS) + n;
    if (vm != 0.0f) {
      int x0 = (int)floorf(xr), y0 = (int)floorf(yr);
      int x1 = x0 + 1,          y1 = y0 + 1;
      float wx = xr - (float)x0, wy = yr - (float)y0;
      float b00 = ((x0 >= 0) && (x0 < FW) && (y0 >= 0) && (y0 < FH)) ? 1.f : 0.f;
      float b01 = ((x1 >= 0) && (x1 < FW) && (y0 >= 0) && (y0 < FH)) ? 1.f : 0.f;
      float b10 = ((x0 >= 0) && (x0 < FW) && (y1 >= 0) && (y1 < FH)) ? 1.f : 0.f;
      float b11 = ((x1 >= 0) && (x1 < FW) && (y1 >= 0) && (y1 < FH)) ? 1.f : 0.f;
      float w00 = (1.f - wx) * (1.f - wy) * b00;
      float w01 = wx * (1.f - wy) * b01;
      float w10 = (1.f - wx) * wy * b10;
      float w11 = wx * wy * b11;
      int x0c = min(max(x0, 0), FW - 1), x1c = min(max(x1, 0), FW - 1);
      int y0c = min(max(y0, 0), FH - 1), y1c = min(max(y1, 0), FH - 1);
      int i00 = y0c * FW + x0c, i01 = y0c * FW + x1c;
      int i10 = y1c * FW + x0c, i11 = y1c * FW + x1c;
      const float* fb = feats + (size_t)cam * NFEAT * FH * FW;
#pragma unroll 4
      for (int c = 0; c < NFEAT; ++c) {
        const float* f = fb + c * (FH * FW);
        float val = f[i00] * w00 + f[i01] * w01 + f[i10] * w10 + f[i11] * w11;
        ob[(size_t)c * NPTS] = val;            // coalesced in n per channel
      }
      ob[(size_t)NFEAT * NPTS] = depth * 0.01f;  // depth / VOXEL_SIZE[0]
    } else {
#pragma unroll 4
      for (int c = 0; c < C1; ++c) ob[(size_t)c * NPTS] = 0.0f;
    }
    out[(size_t)NCAM * C1 * NPTS + (size_t)cam * NPTS + n] = vm;
  }
}

// ---------------------------------------------------------------
// launch
// ---------------------------------------------------------------
extern "C" void kernel_launch(void* const* d_in, const int* in_sizes, int n_in,
                              void* d_out, int out_size, void* d_ws, size_t ws_size,
                              hipStream_t stream) {
  const float* mask  = (const float*)d_in[0];   // (1,6,1,352,640)
  const float* intr  = (const float*)d_in[1];   // (1,6,3,3)
  const float* extr  = (const float*)d_in[2];   // (1,6,4,4)
  const float* feats = (const float*)d_in[3];   // (1,6,64,44,80)
  float* out = (float*)d_out;                   // 6*65*200000 + 6*200000 floats

  float* Mats  = (float*)d_ws;                  // 6*12 floats (pad to 128)
  float* maskr = Mats + 128;                    // 6*44*80 floats

  build_mats<<<1, 32, 0, stream>>>(intr, extr, Mats);

  int nmr = NCAM * FH * FW;
  resize_mask<<<(nmr + 255) / 256, 256, 0, stream>>>(mask, maskr);

  dim3 grid((NPTS + 255) / 256, NCAM);
  project_sample<<<grid, 256, 0, stream>>>(Mats, maskr, feats, out);
}